// SimpleGCN_60610578481383
// MI455X (gfx1250) — compile-verified
//
#include <hip/hip_runtime.h>
#include <hip/hip_bf16.h>

// ---------------- problem constants ----------------
#define N_CONS   4000
#define N_VARS   8000
#define N_TOTAL  (N_CONS + N_VARS)      // 12000
#define N_EDGES  100000
#define IN_VAR   9
#define HID      16
#define N_TILES  (N_TOTAL / HID)        // 750 (12000 divisible by 16)

typedef __attribute__((ext_vector_type(2))) float v2f;
typedef __attribute__((ext_vector_type(8))) float v8f;

// ---------------- workspace layout (bytes) ----------------
// X    : N_TOTAL*HID f32      @ 0          (768000)
// AGG  : N_TOTAL*HID f32      @ 768000     (768000)
// H1   : N_TOTAL*HID f32      @ 1536000    (768000)
// AGG2 : N_TOTAL*HID f32      @ 2304000    (768000)
// DEG  : N_TOTAL f32          @ 3072000    (48000)
// UNIQ : N_EDGES u8           @ 3120000    (100000)
// BMAP : N_CONS*N_VARS bits   @ 3220000    (4000000)   total 7,220,000 B
#define OFF_X     0
#define OFF_AGG   768000
#define OFF_H1    1536000
#define OFF_AGG2  2304000
#define OFF_DEG   3072000
#define OFF_UNIQ  3120000
#define OFF_BMAP  3220000

// ---------------- kernel 1: node embeddings ----------------
__global__ void gcn_embed(const float* __restrict__ cons_x,
                          const float* __restrict__ var_x,
                          const float* __restrict__ Wv, const float* __restrict__ bv,
                          const float* __restrict__ Wc, const float* __restrict__ bc,
                          float* __restrict__ X) {
    int idx = blockIdx.x * blockDim.x + threadIdx.x;     // one thread per (node, h)
    if (idx >= N_TOTAL * HID) return;
    int node = idx >> 4;
    int h    = idx & (HID - 1);
    float acc;
    if (node < N_CONS) {
        acc = cons_x[node] * Wc[h] + bc[h];              // Wc is [1,16]
    } else {
        int vn = node - N_CONS;
        acc = bv[h];
        #pragma unroll
        for (int k = 0; k < IN_VAR; ++k)
            acc = fmaf(var_x[vn * IN_VAR + k], Wv[k * HID + h], acc);
    }
    X[idx] = acc;
}

// ---------------- kernel 2: dedupe + degree + scatter layer-1 ----------------
__global__ void gcn_edge_pass1(const int* __restrict__ ei,
                               const float* __restrict__ X,
                               float* __restrict__ AGG,
                               float* __restrict__ DEG,
                               unsigned int* __restrict__ bmap,
                               unsigned char* __restrict__ uniq) {
    int e = blockIdx.x * blockDim.x + threadIdx.x;
    if (e >= N_EDGES) return;
    int c = ei[e];
    int v = ei[N_EDGES + e];
    unsigned long long pos = (unsigned long long)c * (unsigned long long)N_VARS +
                             (unsigned long long)(unsigned)v;
    unsigned int word = (unsigned int)(pos >> 5);
    unsigned int bit  = 1u << ((unsigned int)pos & 31u);
    unsigned int old  = atomicOr(&bmap[word], bit);
    if (old & bit) { uniq[e] = 0; return; }              // duplicate edge: adj stays 1.0
    uniq[e] = 1;
    atomicAdd(&DEG[c], 1.0f);
    atomicAdd(&DEG[N_CONS + v], 1.0f);
    const float* xv = X + (size_t)(N_CONS + v) * HID;
    const float* xc = X + (size_t)c * HID;
    float* ac = AGG + (size_t)c * HID;
    float* av = AGG + (size_t)(N_CONS + v) * HID;
    #pragma unroll
    for (int h = 0; h < HID; ++h) {
        atomicAdd(&ac[h], xv[h]);
        atomicAdd(&av[h], xc[h]);
    }
}

// ---------------- kernel 3: scatter layer-2 (reuse dedup flags) ----------------
__global__ void gcn_edge_pass2(const int* __restrict__ ei,
                               const unsigned char* __restrict__ uniq,
                               const float* __restrict__ H1,
                               float* __restrict__ AGG2) {
    int e = blockIdx.x * blockDim.x + threadIdx.x;
    if (e >= N_EDGES) return;
    if (!uniq[e]) return;
    int c = ei[e];
    int v = ei[N_EDGES + e];
    const float* xv = H1 + (size_t)(N_CONS + v) * HID;
    const float* xc = H1 + (size_t)c * HID;
    float* ac = AGG2 + (size_t)c * HID;
    float* av = AGG2 + (size_t)(N_CONS + v) * HID;
    #pragma unroll
    for (int h = 0; h < HID; ++h) {
        atomicAdd(&ac[h], xv[h]);
        atomicAdd(&av[h], xc[h]);
    }
}

// ---------------- kernel 4: (deg^-1 * AGG) @ W + b via V_WMMA_F32_16X16X4_F32 ----------------
// One wave32 per 16-node tile. K=16 split into 4 chained WMMAs of K=4.
// A 16x4 f32 layout  : lane L (half = L>>4) holds A[L&15][kbase + 2*half + {0,1}]
// C/D 16x16 f32 layout: vgpr r, lane L -> D[r + 8*(L>>4)][L&15]
__global__ void gcn_layer_wmma(const float* __restrict__ AGG,
                               const float* __restrict__ DEG,
                               const float* __restrict__ W,   // [16,16] row-major [in][out]
                               const float* __restrict__ bias,
                               float* __restrict__ OUT,
                               float* __restrict__ pool,      // may be null
                               int do_relu) {
    int wave = threadIdx.x >> 5;
    int lane = threadIdx.x & 31;
    int tile = blockIdx.x * (blockDim.x >> 5) + wave;
    if (tile >= N_TILES) return;          // wave-uniform: EXEC stays all-ones for WMMA

    int half = lane >> 4;
    int mrow = tile * HID + (lane & 15);  // node row this lane contributes to A
    int col  = lane & 15;                 // output column for B / C / D

    float dscale = 1.0f / fmaxf(DEG[mrow], 1.0f);

    v8f acc = {};
    #pragma unroll
    for (int k0 = 0; k0 < 4; ++k0) {
        int ka = 4 * k0 + 2 * half;
        v2f a, b;
        a.x = AGG[(size_t)mrow * HID + ka]     * dscale;
        a.y = AGG[(size_t)mrow * HID + ka + 1] * dscale;
        b.x = W[(size_t)ka * HID + col];
        b.y = W[(size_t)(ka + 1) * HID + col];
        acc = __builtin_amdgcn_wmma_f32_16x16x4_f32(
            /*neg_a=*/false, a, /*neg_b=*/false, b,
            /*c_mod=*/(short)0, acc, /*reuse_a=*/false, /*reuse_b=*/false);
    }

    float bcol = bias[col];
    float lsum = 0.0f;
    #pragma unroll
    for (int r = 0; r < 8; ++r) {
        int orow = tile * HID + r + 8 * half;
        float v = acc[r] + bcol;
        if (do_relu) v = fmaxf(v, 0.0f);
        OUT[(size_t)orow * HID + col] = v;
        lsum += v;
    }
    if (pool) atomicAdd(&pool[col], lsum * (1.0f / (float)N_TOTAL));
}

// ---------------- launcher ----------------
extern "C" void kernel_launch(void* const* d_in, const int* in_sizes, int n_in,
                              void* d_out, int out_size, void* d_ws, size_t ws_size,
                              hipStream_t stream) {
    const float* cons_x = (const float*)d_in[0];
    const float* var_x  = (const float*)d_in[1];
    const int*   ei     = (const int*)  d_in[2];
    // d_in[3] = edge_attr (unused by the reference)
    const float* W_var  = (const float*)d_in[4];
    const float* b_var  = (const float*)d_in[5];
    const float* W_cons = (const float*)d_in[6];
    const float* b_cons = (const float*)d_in[7];
    const float* W1     = (const float*)d_in[8];
    const float* b1     = (const float*)d_in[9];
    const float* W2     = (const float*)d_in[10];
    const float* b2     = (const float*)d_in[11];
    float* out = (float*)d_out;

    char* ws = (char*)d_ws;
    float*         X    = (float*)        (ws + OFF_X);
    float*         AGG  = (float*)        (ws + OFF_AGG);
    float*         H1   = (float*)        (ws + OFF_H1);
    float*         AGG2 = (float*)        (ws + OFF_AGG2);
    float*         DEG  = (float*)        (ws + OFF_DEG);
    unsigned char* UNIQ = (unsigned char*)(ws + OFF_UNIQ);
    unsigned int*  BMAP = (unsigned int*) (ws + OFF_BMAP);

    // zero AGG..DEG span (AGG, H1, AGG2, DEG contiguous), the dedup bitmap, and d_out
    hipMemsetAsync(ws + OFF_AGG, 0, (size_t)(OFF_UNIQ - OFF_AGG), stream);
    hipMemsetAsync(ws + OFF_BMAP, 0, (size_t)(N_CONS) * N_VARS / 8, stream);
    hipMemsetAsync(d_out, 0, (size_t)HID * sizeof(float), stream);

    // 1) embeddings
    {
        int n = N_TOTAL * HID;
        gcn_embed<<<(n + 255) / 256, 256, 0, stream>>>(cons_x, var_x, W_var, b_var,
                                                       W_cons, b_cons, X);
    }
    // 2) dedupe + degree + layer-1 aggregation
    gcn_edge_pass1<<<(N_EDGES + 255) / 256, 256, 0, stream>>>(ei, X, AGG, DEG, BMAP, UNIQ);
    // 3) h1 = relu((D^-1 A x) @ W1 + b1)
    {
        int waves_per_block = 8;                     // 256 threads
        int blocks = (N_TILES + waves_per_block - 1) / waves_per_block;
        gcn_layer_wmma<<<blocks, 32 * waves_per_block, 0, stream>>>(
            AGG, DEG, W1, b1, H1, nullptr, /*relu=*/1);
    }
    // 4) layer-2 aggregation
    gcn_edge_pass2<<<(N_EDGES + 255) / 256, 256, 0, stream>>>(ei, UNIQ, H1, AGG2);
    // 5) x2 = (D^-1 A h1) @ W2 + b2, fused mean-pool into d_out (reuse X as output buf)
    {
        int waves_per_block = 8;
        int blocks = (N_TILES + waves_per_block - 1) / waves_per_block;
        gcn_layer_wmma<<<blocks, 32 * waves_per_block, 0, stream>>>(
            AGG2, DEG, W2, b2, X, out, /*relu=*/0);
    }
}